// MotionModel_87574383165964
// MI455X (gfx1250) — compile-verified
//
#include <hip/hip_runtime.h>
#include <hip/hip_bf16.h>

typedef __attribute__((ext_vector_type(2))) float v2f;
typedef __attribute__((ext_vector_type(4))) float v4f;
typedef __attribute__((ext_vector_type(8))) float v8f;

#define LOG2E      1.4426950408889634f
#define TWO_LOG2E  2.8853900817779268f

__device__ __forceinline__ float fast_tanh(float x) {
    // tanh(x) = 1 - 2/(exp(2x)+1), exp via hardware exp2
    float t = __builtin_amdgcn_exp2f(x * TWO_LOG2E);
    return 1.0f - 2.0f * __builtin_amdgcn_rcpf(t + 1.0f);
}

__device__ __forceinline__ void lds_fence() {
    asm volatile("s_wait_dscnt 0" ::: "memory");
}

// D = A(16x4, f32) * B(4x16, f32) + C(16x16, f32)
__device__ __forceinline__ v8f wmma4(v2f a, v2f b, v8f c) {
    return __builtin_amdgcn_wmma_f32_16x16x4_f32(
        /*neg_a=*/false, a, /*neg_b=*/false, b,
        /*c_mod=*/(short)0, c, /*reuse_a=*/false, /*reuse_b=*/false);
}

// Per-wave LDS tile: 32 particles x 20 floats (stride 20 keeps b64/b128 alignment,
// and 20*4B stride gives conflict-free bank access patterns).
#define ROWSTRIDE 20

__global__ __launch_bounds__(256)
void motion_model_kernel(const float* __restrict__ ps,
                         const float* __restrict__ eps,
                         const float* __restrict__ W1, const float* __restrict__ b1,
                         const float* __restrict__ W2, const float* __restrict__ b2,
                         const float* __restrict__ W3, const float* __restrict__ b3,
                         float* __restrict__ out_sample,
                         float* __restrict__ out_prec,
                         int NM)
{
    __shared__ float hbuf[8][32 * ROWSTRIDE];

    const int tid  = threadIdx.x;
    const int wave = tid >> 5;
    const int lane = tid & 31;
    const int n    = lane & 15;   // column within 16x16 tile
    const int kh   = lane >> 4;   // which K/M half this lane holds
    float* hb = hbuf[wave];

    // ---- Preload B fragments (weights as 4x16 / K-sliced B operands) ----
    // B-layout (32-bit, KxN=4x16): VGPR0 = K=0 (lanes 0-15) / K=2 (lanes 16-31),
    //                              VGPR1 = K=1 / K=3.
    const bool n8  = (n < 8);
    const bool n14 = (n < 14);

    v2f B1f;   // W1: (4,8) padded to 4x16
    B1f.x = n8 ? W1[(2 * kh + 0) * 8 + n] : 0.0f;
    B1f.y = n8 ? W1[(2 * kh + 1) * 8 + n] : 0.0f;

    v2f B2a, B2b;  // W2: (8,16), split K = 0..3 and 4..7
    B2a.x = W2[(0 + 2 * kh) * 16 + n];
    B2a.y = W2[(1 + 2 * kh) * 16 + n];
    B2b.x = W2[(4 + 2 * kh) * 16 + n];
    B2b.y = W2[(5 + 2 * kh) * 16 + n];

    v2f B3f[4];    // W3: (16,14) padded to 16x16, K split in 4
#pragma unroll
    for (int j = 0; j < 4; ++j) {
        B3f[j].x = n14 ? W3[(4 * j + 0 + 2 * kh) * 14 + n] : 0.0f;
        B3f[j].y = n14 ? W3[(4 * j + 1 + 2 * kh) * 14 + n] : 0.0f;
    }

    const float b1v = n8  ? b1[n] : 0.0f;
    const float b2v = b2[n];
    const float b3v = n14 ? b3[n] : 0.0f;

    const long long base = (long long)blockIdx.x * 256 + (long long)wave * 32;

    // ---- GEMM phase: two 16-particle tiles per wave ----
#pragma unroll
    for (int t = 0; t < 2; ++t) {
        const long long p0 = base + t * 16;
        long long pc = p0 + n;
        if (pc >= NM) pc = NM - 1;

        // A1 (16x4): lane holds particle n, K = 2*kh .. 2*kh+1  -> one b64 load
        v2f a1 = *(const v2f*)(ps + pc * 4 + 2 * kh);

        // Layer 1: 16x4 @ 4x16
        v8f c1 = {};
        c1 = wmma4(a1, B1f, c1);

        const int crow = (t * 16 + kh * 8) * ROWSTRIDE + n;
#pragma unroll
        for (int r = 0; r < 8; ++r)
            hb[crow + r * ROWSTRIDE] = fast_tanh(c1[r] + b1v);
        lds_fence();

        // A fragments for this tile: particle row (t*16+n), K base + 2*kh
        const int arow = (t * 16 + n) * ROWSTRIDE + 2 * kh;

        // Layer 2: 16x8 @ 8x16 (two chained WMMAs)
        v2f a2a = *(const v2f*)(hb + arow + 0);
        v2f a2b = *(const v2f*)(hb + arow + 4);
        v8f c2 = {};
        c2 = wmma4(a2a, B2a, c2);
        c2 = wmma4(a2b, B2b, c2);

#pragma unroll
        for (int r = 0; r < 8; ++r)
            hb[crow + r * ROWSTRIDE] = fast_tanh(c2[r] + b2v);
        lds_fence();

        // Layer 3: 16x16 @ 16x14 (four chained WMMAs)
        v2f a30 = *(const v2f*)(hb + arow + 0);
        v2f a31 = *(const v2f*)(hb + arow + 4);
        v2f a32 = *(const v2f*)(hb + arow + 8);
        v2f a33 = *(const v2f*)(hb + arow + 12);
        v8f c3 = {};
        c3 = wmma4(a30, B3f[0], c3);
        c3 = wmma4(a31, B3f[1], c3);
        c3 = wmma4(a32, B3f[2], c3);
        c3 = wmma4(a33, B3f[3], c3);

#pragma unroll
        for (int r = 0; r < 8; ++r)
            hb[crow + r * ROWSTRIDE] = c3[r] + b3v;
        lds_fence();
    }

    // ---- Per-particle phase: lane <-> particle (base + lane) ----
    const long long p  = base + lane;
    const bool valid   = (p < NM);
    const long long pl = valid ? p : (long long)(NM - 1);

    const float* xr = hb + lane * ROWSTRIDE;
    v4f mean = *(const v4f*)(xr + 0);   // x[0..3]
    v4f ldg  = *(const v4f*)(xr + 4);   // x[4..7]  log-diag
    v4f low0 = *(const v4f*)(xr + 8);   // x[8..11] l10 l20 l21 l30
    v2f low1 = *(const v2f*)(xr + 12);  // x[12..13] l31 l32

    v4f e = *(const v4f*)(eps + pl * 4);

    const float l10 = low0.x, l20 = low0.y, l21 = low0.z;
    const float l30 = low0.w, l31 = low1.x, l32 = low1.y;

    // diag = exp(clip(log_diag, -2, 1))
    float d0 = __builtin_amdgcn_exp2f(fminf(fmaxf(ldg.x, -2.0f), 1.0f) * LOG2E);
    float d1 = __builtin_amdgcn_exp2f(fminf(fmaxf(ldg.y, -2.0f), 1.0f) * LOG2E);
    float d2 = __builtin_amdgcn_exp2f(fminf(fmaxf(ldg.z, -2.0f), 1.0f) * LOG2E);
    float d3 = __builtin_amdgcn_exp2f(fminf(fmaxf(ldg.w, -2.0f), 1.0f) * LOG2E);

    // sample = mean + L @ eps
    v4f s;
    s.x = mean.x + d0 * e.x;
    s.y = mean.y + l10 * e.x + d1 * e.y;
    s.z = mean.z + l20 * e.x + l21 * e.y + d2 * e.z;
    s.w = mean.w + l30 * e.x + l31 * e.y + l32 * e.z + d3 * e.w;

    // Linv (forward substitution of L X = I)
    const float r0 = 1.0f / d0, r1 = 1.0f / d1, r2 = 1.0f / d2, r3 = 1.0f / d3;
    const float i00 = r0;
    const float i11 = r1;
    const float i10 = -l10 * i00 * r1;
    const float i22 = r2;
    const float i21 = -(l21 * i11) * r2;
    const float i20 = -(l20 * i00 + l21 * i10) * r2;
    const float i33 = r3;
    const float i32 = -(l32 * i22) * r3;
    const float i31 = -(l31 * i11 + l32 * i21) * r3;
    const float i30 = -(l30 * i00 + l31 * i10 + l32 * i20) * r3;

    // precision = Linv^T @ Linv (symmetric)
    const float P00 = i00 * i00 + i10 * i10 + i20 * i20 + i30 * i30;
    const float P01 = i10 * i11 + i20 * i21 + i30 * i31;
    const float P02 = i20 * i22 + i30 * i32;
    const float P03 = i30 * i33;
    const float P11 = i11 * i11 + i21 * i21 + i31 * i31;
    const float P12 = i21 * i22 + i31 * i32;
    const float P13 = i31 * i33;
    const float P22 = i22 * i22 + i32 * i32;
    const float P23 = i32 * i33;
    const float P33 = i33 * i33;

    if (valid) {
        *(v4f*)(out_sample + p * 4) = s;
        float* pp = out_prec + p * 16;
        *(v4f*)(pp + 0)  = (v4f){P00, P01, P02, P03};
        *(v4f*)(pp + 4)  = (v4f){P01, P11, P12, P13};
        *(v4f*)(pp + 8)  = (v4f){P02, P12, P22, P23};
        *(v4f*)(pp + 12) = (v4f){P03, P13, P23, P33};
    }
}

extern "C" void kernel_launch(void* const* d_in, const int* in_sizes, int n_in,
                              void* d_out, int out_size, void* d_ws, size_t ws_size,
                              hipStream_t stream) {
    const float* ps  = (const float*)d_in[0];
    const float* eps = (const float*)d_in[1];
    const float* W1  = (const float*)d_in[2];
    const float* b1  = (const float*)d_in[3];
    const float* W2  = (const float*)d_in[4];
    const float* b2  = (const float*)d_in[5];
    const float* W3  = (const float*)d_in[6];
    const float* b3  = (const float*)d_in[7];

    const int NM = in_sizes[0] / 4;            // N*M particles (D=4)
    float* out_sample = (float*)d_out;         // N*M*4
    float* out_prec   = out_sample + (size_t)NM * 4;  // N*M*16

    const int block = 256;                     // 8 waves; 32 particles/wave
    const int grid  = (NM + block - 1) / block;
    motion_model_kernel<<<grid, block, 0, stream>>>(
        ps, eps, W1, b1, W2, b2, W3, b3, out_sample, out_prec, NM);
}